// UnarySqrt_47519518163294
// MI455X (gfx1250) — compile-verified
//
#include <hip/hip_runtime.h>
#include <hip/hip_bf16.h>

// UnarySqrt bitstream scan: L=64 steps, N columns independent.
// Memory-bound stream (read 256MB + write 256MB) -> ~22us floor @ 23.3 TB/s.
// CDNA5 path: 4-deep ASYNCcnt ring of global_load_async_to_lds_b128 per wave,
// ds_load_b128 consume, NT b128 stores, global_prefetch_b8 ahead.

typedef float v4f __attribute__((ext_vector_type(4)));
typedef unsigned int v4u __attribute__((ext_vector_type(4)));

#define STEPS 64
#define VEC 4
#define BLOCK 256
#define RING 4          // async ring depth (slots per wave)
#define SLOT_BYTES 512  // 32 lanes * 16B
#define PF_DIST 8

// One bit-step for 4 packed state machines. Input bits are 0x0 / 0x3f800000.
__device__ __forceinline__ v4f step4(v4u bits, int cnt[VEC], int sr[VEC],
                                     int acc[VEC]) {
    v4f o;
#pragma unroll
    for (int k = 0; k < VEC; ++k) {
        int x       = (bits[k] != 0u) ? 1 : 0;
        int emit_en = (x ^ 1) & ((acc[k] > 0) ? 1 : 0);
        int ob      = x | emit_en;                     // output bit
        // divisor==1 -> dividend_sync (= old cnt!=0, dividend==0); else sr
        int q = ob ? ((cnt[k] != 0) ? 1 : 0) : sr[k];
        cnt[k] = ob ? (cnt[k] - ((cnt[k] != 0) ? 1 : 0))
                    : (cnt[k] + ((cnt[k] != 3) ? 1 : 0));
        sr[k] = q;
        int a = acc[k] + q - emit_en;                  // lower clamp never binds
        acc[k] = (a > 7) ? 7 : a;
        o[k] = ob ? 1.0f : 0.0f;
    }
    return o;
}

#define ASYNC_LOAD_B128(ldsaddr_u32, gptr)                                   \
    asm volatile("global_load_async_to_lds_b128 %0, %1, off"                 \
                 :: "v"(ldsaddr_u32), "v"(gptr) : "memory")

__global__ __launch_bounds__(BLOCK) void sqrt_bitstream_kernel(
    const float* __restrict__ in, float* __restrict__ out, int ncols)
{
    __shared__ __align__(16) unsigned char stage[(BLOCK / 32) * RING * SLOT_BYTES];

    const int lane = threadIdx.x & 31;
    const int wave = threadIdx.x >> 5;

    const long long col0 =
        ((long long)blockIdx.x * BLOCK + threadIdx.x) * (long long)VEC;
    if (col0 >= ncols) return;

    const long long stride = ncols;

    if (col0 + VEC <= (long long)ncols) {
        int cnt[VEC], sr[VEC], acc[VEC];
#pragma unroll
        for (int k = 0; k < VEC; ++k) { cnt[k] = 0; sr[k] = 1; acc[k] = 0; }

        const float* ip = in + col0;
        float* op = out + col0;

        // Per-wave ring slots: generic ptrs (for ds_load) + LDS byte addresses
        // (low 32 bits of a generic LDS pointer == LDS offset).
        unsigned char* sp[RING];
        unsigned sa[RING];
#pragma unroll
        for (int s = 0; s < RING; ++s) {
            sp[s] = &stage[((wave * RING + s) * SLOT_BYTES) + (lane << 4)];
            sa[s] = (unsigned)(size_t)sp[s];
        }

        // ---- prologue: fill the async ring with steps 0..RING-1 ----
#pragma unroll
        for (int s = 0; s < RING; ++s) {
            const float* g = ip + (long long)s * stride;
            ASYNC_LOAD_B128(sa[s], g);
        }

        // ---- steady state: consume step l, reissue slot for step l+RING ----
#pragma unroll 4
        for (int l = 0; l < STEPS - RING; ++l) {
            const int s = l & (RING - 1);
            // RING loads outstanding; <=RING-1 left => oldest (step l) landed
            asm volatile("s_wait_asynccnt 0x3" ::: "memory");
            v4u bits = *(const v4u*)sp[s];
            v4f o = step4(bits, cnt, sr, acc);
            __builtin_nontemporal_store(o, (v4f*)(op + (long long)l * stride));
            // ds read must retire before async engine overwrites the slot
            const float* g = ip + (long long)(l + RING) * stride;
            asm volatile("s_wait_dscnt 0x0" ::: "memory");
            ASYNC_LOAD_B128(sa[s], g);
            if (l + PF_DIST < STEPS)
                __builtin_prefetch(ip + (long long)(l + PF_DIST) * stride, 0, 0);
        }

        // ---- epilogue: no new issues; must wait all remaining to zero ----
#pragma unroll
        for (int l = STEPS - RING; l < STEPS; ++l) {
            const int s = l & (RING - 1);
            asm volatile("s_wait_asynccnt 0x0" ::: "memory");
            v4u bits = *(const v4u*)sp[s];
            v4f o = step4(bits, cnt, sr, acc);
            __builtin_nontemporal_store(o, (v4f*)(op + (long long)l * stride));
        }
    } else {
        // ---- scalar tail (ncols % 4 != 0; not hit for N=2^20) ----
        for (long long c = col0; c < (long long)ncols; ++c) {
            int cnt = 0, sr = 1, acc = 0;
            for (int l = 0; l < STEPS; ++l) {
                unsigned bu = __builtin_nontemporal_load(
                    (const unsigned*)(in + (long long)l * stride + c));
                int x       = (bu != 0u) ? 1 : 0;
                int emit_en = (x ^ 1) & ((acc > 0) ? 1 : 0);
                int ob      = x | emit_en;
                int q       = ob ? ((cnt != 0) ? 1 : 0) : sr;
                cnt = ob ? (cnt - ((cnt != 0) ? 1 : 0))
                         : (cnt + ((cnt != 3) ? 1 : 0));
                sr = q;
                int a = acc + q - emit_en;
                acc = (a > 7) ? 7 : a;
                float ov = ob ? 1.0f : 0.0f;
                __builtin_nontemporal_store(ov, out + (long long)l * stride + c);
            }
        }
    }
}

extern "C" void kernel_launch(void* const* d_in, const int* in_sizes, int n_in,
                              void* d_out, int out_size, void* d_ws, size_t ws_size,
                              hipStream_t stream) {
    const float* in = (const float*)d_in[0];
    float* out = (float*)d_out;
    const int total = in_sizes[0];          // L * N
    const int ncols = total / STEPS;        // N
    const int threads = (ncols + VEC - 1) / VEC;
    const int blocks = (threads + BLOCK - 1) / BLOCK;
    sqrt_bitstream_kernel<<<blocks, BLOCK, 0, stream>>>(in, out, ncols);
}